// ESKF_Torch_412316860593
// MI455X (gfx1250) — compile-verified
//
#include <hip/hip_runtime.h>
#include <cstdint>

// ESKF update, one element per thread. Memory-bound: ~592B/elem, ~155MB total
// => ~6.7us floor at 23.3 TB/s. Compute: few hundred f32 FLOPs/elem after
// exploiting H = [[I3,0],[0,Qd]] structure and replacing inv(S) with a solve.
// CDNA5 paths: coalesced global_load_async_to_lds_b128 staging (ASYNCcnt) of
// AoS cov/Q/R, s_wait_asynccnt, per-lane ds_load reads, and coalesced
// global_store_async_from_lds_b128 for the output. Full blocks use a fully
// unrolled copy sequence exploiting that IOFFSET offsets BOTH the global and
// the LDS address (ISA 15.18.3), so one VGPR address + immediate offsets.

#define TPB 128
#define DT_F 0.01f

__device__ __forceinline__ void async_b128(uint32_t lds_off, const void* gaddr) {
  asm volatile("global_load_async_to_lds_b128 %0, %1, off"
               :: "v"(lds_off), "v"(gaddr) : "memory");
}
__device__ __forceinline__ void async_b32(uint32_t lds_off, const void* gaddr) {
  asm volatile("global_load_async_to_lds_b32 %0, %1, off"
               :: "v"(lds_off), "v"(gaddr) : "memory");
}
template <int OFF>
__device__ __forceinline__ void async_b128_off(uint32_t lds_off, const void* gaddr) {
  asm volatile("global_load_async_to_lds_b128 %0, %1, off offset:%c2"
               :: "v"(lds_off), "v"(gaddr), "i"(OFF) : "memory");
}
template <int OFF>
__device__ __forceinline__ void async_store_b128_off(const void* gaddr, uint32_t lds_off) {
  asm volatile("global_store_async_from_lds_b128 %0, %1, off offset:%c2"
               :: "v"(gaddr), "v"(lds_off), "i"(OFF) : "memory");
}
__device__ __forceinline__ void wait_async0() {
  asm volatile("s_wait_asynccnt 0" ::: "memory");
}

__device__ __forceinline__ void qmul(const float a[4], const float b[4], float o[4]) {
  o[0] = a[0]*b[0] - a[1]*b[1] - a[2]*b[2] - a[3]*b[3];
  o[1] = a[0]*b[1] + a[1]*b[0] + a[2]*b[3] - a[3]*b[2];
  o[2] = a[0]*b[2] - a[1]*b[3] + a[2]*b[0] + a[3]*b[1];
  o[3] = a[0]*b[3] + a[1]*b[2] - a[2]*b[1] + a[3]*b[0];
}

__global__ __launch_bounds__(TPB)
void eskf_kernel(const float* __restrict__ x, const float* __restrict__ state,
                 const float* __restrict__ cov, const float* __restrict__ Qm,
                 const float* __restrict__ Rm, float* __restrict__ out, int Bv)
{
  // LDS floats: cov [0,36*TPB) | Q [36*TPB,72*TPB) | R [72*TPB,121*TPB)
  // Output staging reuses [0, 7*TPB) after cov is consumed.
  __shared__ float smem[TPB * 121];            // 61,952 bytes
  const int tid   = threadIdx.x;
  const int elem0 = blockIdx.x * TPB;
  const int i     = elem0 + tid;
  const int nE    = min(TPB, Bv - elem0);
  const bool full = (nE == TPB);

  const uint32_t sbase = (uint32_t)(uintptr_t)(&smem[0]); // LDS byte offset
  const uint32_t qB    = sbase + TPB * 36 * 4;
  const uint32_t rB    = sbase + TPB * 72 * 4;

  // ---- issue all async staged copies up front (fully coalesced b128) ----
  if (full) {
    // Fast path: one address per array, immediate offsets advance both sides.
    const char* gc = (const char*)cov + (size_t)elem0 * 144 + (size_t)tid * 16;
    const char* gq = (const char*)Qm  + (size_t)elem0 * 144 + (size_t)tid * 16;
    const char* gr = (const char*)Rm  + (size_t)elem0 * 196 + (size_t)tid * 16;
    const uint32_t lc = sbase + tid * 16;
    const uint32_t lq = qB    + tid * 16;
    const uint32_t lr = rB    + tid * 16;
    async_b128_off<0>(lc, gc);        async_b128_off<2048>(lc, gc);
    async_b128_off<4096>(lc, gc);     async_b128_off<6144>(lc, gc);
    async_b128_off<8192>(lc, gc);     async_b128_off<10240>(lc, gc);
    async_b128_off<12288>(lc, gc);    async_b128_off<14336>(lc, gc);
    async_b128_off<16384>(lc, gc);                       // 9 x 2KB = 36*TPB*4
    async_b128_off<0>(lq, gq);        async_b128_off<2048>(lq, gq);
    async_b128_off<4096>(lq, gq);     async_b128_off<6144>(lq, gq);
    async_b128_off<8192>(lq, gq);     async_b128_off<10240>(lq, gq);
    async_b128_off<12288>(lq, gq);    async_b128_off<14336>(lq, gq);
    async_b128_off<16384>(lq, gq);
    async_b128_off<0>(lr, gr);        async_b128_off<2048>(lr, gr);
    async_b128_off<4096>(lr, gr);     async_b128_off<6144>(lr, gr);
    async_b128_off<8192>(lr, gr);     async_b128_off<10240>(lr, gr);
    async_b128_off<12288>(lr, gr);    async_b128_off<14336>(lr, gr);
    async_b128_off<16384>(lr, gr);    async_b128_off<18432>(lr, gr);
    async_b128_off<20480>(lr, gr);    async_b128_off<22528>(lr, gr);
    if (tid < 32) {                   // 1568 = 12*128 + 32 float4s
      const char* grt = (const char*)Rm + (size_t)elem0 * 196 + (size_t)(1536 + tid) * 16;
      async_b128(rB + (uint32_t)(1536 + tid) * 16, grt);
    }
  } else {
    const float4* gc = (const float4*)(cov + (size_t)elem0 * 36);
    const float4* gq = (const float4*)(Qm  + (size_t)elem0 * 36);
    const int nf4 = (nE * 36) >> 2;
    for (int k = tid; k < nf4; k += TPB) {
      async_b128(sbase + (uint32_t)k * 16, gc + k);
      async_b128(qB    + (uint32_t)k * 16, gq + k);
    }
    const float* gr = Rm + (size_t)elem0 * 49;
    const int rw  = nE * 49;
    const int rf4 = rw >> 2;
    for (int k = tid; k < rf4; k += TPB)
      async_b128(rB + (uint32_t)k * 16, (const float4*)gr + k);
    for (int k = rf4 * 4 + tid; k < rw; k += TPB)
      async_b32(rB + (uint32_t)k * 4, gr + k);
  }

  // ---- overlap: direct coalesced loads of x rows and state ----
  const int ic = (i < Bv) ? i : (Bv - 1);
  float m[7], tw[6], st[7];
  #pragma unroll
  for (int j = 0; j < 7; ++j) m[j]  = x[(size_t)j * Bv + ic];
  #pragma unroll
  for (int j = 0; j < 6; ++j) tw[j] = x[(size_t)(8 + j) * Bv + ic];
  #pragma unroll
  for (int j = 0; j < 7; ++j) st[j] = state[(size_t)ic * 7 + j];

  wait_async0();      // our wave's LDS writes landed (ASYNCcnt==0)
  __syncthreads();    // cross-wave visibility

  float o0 = 0.f, o1 = 0.f, o2 = 0.f, o3 = 0.f, o4 = 0.f, o5 = 0.f, o6 = 0.f;
  if (i < Bv) {
    // ---- P = cov + Q (6x6, registers) ----
    float P[6][6];
    #pragma unroll
    for (int r = 0; r < 6; ++r)
      #pragma unroll
      for (int c = 0; c < 6; ++c)
        P[r][c] = smem[tid * 36 + r * 6 + c] + smem[TPB * 36 + tid * 36 + r * 6 + c];

    const float* rS = &smem[TPB * 72 + tid * 49]; // this element's R (7x7)

    // ---- predict: inject(state, DT*twist) ----
    float pp[3] = { st[0] + DT_F * tw[0], st[1] + DT_F * tw[1], st[2] + DT_F * tw[2] };
    float q0[4] = { st[3], st[4], st[5], st[6] };
    float dq[4] = { 1.0f, 0.5f * DT_F * tw[3], 0.5f * DT_F * tw[4], 0.5f * DT_F * tw[5] };
    float qp[4]; qmul(q0, dq, qp);
    {
      float n = 1.0f / sqrtf(qp[0]*qp[0] + qp[1]*qp[1] + qp[2]*qp[2] + qp[3]*qp[3]);
      qp[0] *= n; qp[1] *= n; qp[2] *= n; qp[3] *= n;
    }

    // ---- G = Qd(q0): 0.5 * [[-x,-y,-z],[w,-z,y],[z,w,-x],[-y,x,w]] ----
    const float G[4][3] = {
      { -0.5f*q0[1], -0.5f*q0[2], -0.5f*q0[3] },
      {  0.5f*q0[0], -0.5f*q0[3],  0.5f*q0[2] },
      {  0.5f*q0[3],  0.5f*q0[0], -0.5f*q0[1] },
      { -0.5f*q0[2],  0.5f*q0[1],  0.5f*q0[0] } };

    // ---- S = H P H^T + R with H = [[I3,0],[0,G]] ----
    float S[7][7];
    #pragma unroll
    for (int r = 0; r < 3; ++r)
      #pragma unroll
      for (int c = 0; c < 3; ++c)
        S[r][c] = P[r][c] + rS[r * 7 + c];
    #pragma unroll
    for (int r = 0; r < 3; ++r)                // Bm*G^T  (3x4)
      #pragma unroll
      for (int c = 0; c < 4; ++c) {
        float acc = rS[r * 7 + 3 + c];
        #pragma unroll
        for (int k = 0; k < 3; ++k) acc += P[r][3 + k] * G[c][k];
        S[r][3 + c] = acc;
      }
    #pragma unroll
    for (int r = 0; r < 4; ++r)                // G*C  (4x3)
      #pragma unroll
      for (int c = 0; c < 3; ++c) {
        float acc = rS[(3 + r) * 7 + c];
        #pragma unroll
        for (int k = 0; k < 3; ++k) acc += G[r][k] * P[3 + k][c];
        S[3 + r][c] = acc;
      }
    float GD[4][3];                            // G*D
    #pragma unroll
    for (int r = 0; r < 4; ++r)
      #pragma unroll
      for (int c = 0; c < 3; ++c) {
        float acc = 0.0f;
        #pragma unroll
        for (int k = 0; k < 3; ++k) acc += G[r][k] * P[3 + k][3 + c];
        GD[r][c] = acc;
      }
    #pragma unroll
    for (int r = 0; r < 4; ++r)                // (G*D)*G^T + R  (4x4)
      #pragma unroll
      for (int c = 0; c < 4; ++c) {
        float acc = rS[(3 + r) * 7 + 3 + c];
        #pragma unroll
        for (int k = 0; k < 3; ++k) acc += GD[r][k] * G[c][k];
        S[3 + r][3 + c] = acc;
      }

    // ---- y = S^{-1} (measurement - predict)  (Gaussian elim, S is SPD) ----
    float y[7] = { m[0] - pp[0], m[1] - pp[1], m[2] - pp[2],
                   m[3] - qp[0], m[4] - qp[1], m[5] - qp[2], m[6] - qp[3] };
    #pragma unroll
    for (int c = 0; c < 7; ++c) {
      float inv = 1.0f / S[c][c];
      #pragma unroll
      for (int r = 0; r < 7; ++r)
        if (r > c) {
          float f = S[r][c] * inv;
          #pragma unroll
          for (int k = 0; k < 7; ++k)
            if (k > c) S[r][k] -= f * S[c][k];
          y[r] -= f * y[c];
        }
    }
    #pragma unroll
    for (int c = 6; c >= 0; --c) {
      float acc = y[c];
      #pragma unroll
      for (int k = 0; k < 7; ++k)
        if (k > c) acc -= S[c][k] * y[k];
      y[c] = acc / S[c][c];
    }

    // ---- error_state = P * (H^T y) ----
    float hty[6] = { y[0], y[1], y[2],
      G[0][0]*y[3] + G[1][0]*y[4] + G[2][0]*y[5] + G[3][0]*y[6],
      G[0][1]*y[3] + G[1][1]*y[4] + G[2][1]*y[5] + G[3][1]*y[6],
      G[0][2]*y[3] + G[1][2]*y[4] + G[2][2]*y[5] + G[3][2]*y[6] };
    float es[6];
    #pragma unroll
    for (int r = 0; r < 6; ++r) {
      float acc = 0.0f;
      #pragma unroll
      for (int c = 0; c < 6; ++c) acc += P[r][c] * hty[c];
      es[r] = acc;
    }

    // ---- new_state = inject(predict_state, error_state) ----
    float dq2[4] = { 1.0f, 0.5f * es[3], 0.5f * es[4], 0.5f * es[5] };
    float qf[4]; qmul(qp, dq2, qf);
    float n2 = 1.0f / sqrtf(qf[0]*qf[0] + qf[1]*qf[1] + qf[2]*qf[2] + qf[3]*qf[3]);

    o0 = pp[0] + es[0];
    o1 = pp[1] + es[1];
    o2 = pp[2] + es[2];
    o3 = qf[0] * n2;
    o4 = qf[1] * n2;
    o5 = qf[2] * n2;
    o6 = qf[3] * n2;
  }

  // ---- coalesced output via LDS staging + async store (full blocks) ----
  if (full) {
    __syncthreads();                 // all LDS reads done; safe to reuse region
    float* stg = &smem[tid * 7];
    stg[0] = o0; stg[1] = o1; stg[2] = o2; stg[3] = o3;
    stg[4] = o4; stg[5] = o5; stg[6] = o6;
    __syncthreads();                 // stage complete
    const char* go = (const char*)out + (size_t)elem0 * 28 + (size_t)tid * 16;
    const uint32_t lo = sbase + tid * 16;
    async_store_b128_off<0>(go, lo);           // 224 = 128 + 96 float4s
    if (tid < 96) async_store_b128_off<2048>(go, lo);
    // s_endpgm performs an implicit wait-idle covering outstanding ASYNCcnt.
  } else if (i < Bv) {
    float* o = out + (size_t)i * 7;
    o[0] = o0; o[1] = o1; o[2] = o2; o[3] = o3; o[4] = o4; o[5] = o5; o[6] = o6;
  }
}

extern "C" void kernel_launch(void* const* d_in, const int* in_sizes, int n_in,
                              void* d_out, int out_size, void* d_ws, size_t ws_size,
                              hipStream_t stream) {
  const float* x   = (const float*)d_in[0];   // (14, B)
  const float* st  = (const float*)d_in[1];   // (B, 7, 1)
  const float* cov = (const float*)d_in[2];   // (B, 6, 6)
  const float* Qm  = (const float*)d_in[3];   // (B, 6, 6)
  const float* Rm  = (const float*)d_in[4];   // (B, 7, 7)
  float* out = (float*)d_out;                 // (B, 7, 1)
  const int Bv = in_sizes[0] / 14;
  const int grid = (Bv + TPB - 1) / TPB;
  hipLaunchKernelGGL(eskf_kernel, dim3(grid), dim3(TPB), 0, stream,
                     x, st, cov, Qm, Rm, out, Bv);
}